// GraphAttn_47184510714487
// MI455X (gfx1250) — compile-verified
//
#include <hip/hip_runtime.h>

typedef float v2f __attribute__((ext_vector_type(2)));
typedef float v4f __attribute__((ext_vector_type(4)));
typedef float v8f __attribute__((ext_vector_type(8)));
typedef int   v4i __attribute__((ext_vector_type(4)));

#define ALPHA 0.5f
#define BETA  0.2f
#define B_    4
#define N_    4096
#define F_    256
#define ROWS_TOTAL (B_ * N_)
#define CHUNK_M   16                       // H rows per LDS buffer (16*256*4 = 16 KB)
#define NCHUNK    (N_ / CHUNK_M)
#define LOADS_PER_THREAD ((CHUNK_M * F_) / (256 * 4))   // 4 async b128 per thread/chunk

#define AS1 __attribute__((address_space(1)))
#define AS3 __attribute__((address_space(3)))

#if __has_builtin(__builtin_amdgcn_global_load_async_to_lds_b128)
#define HAS_ASYNC_LDS 1
#else
#define HAS_ASYNC_LDS 0
#endif

static __device__ __forceinline__ void async_g2l_b128(const float* g, float* l) {
#if HAS_ASYNC_LDS
    __builtin_amdgcn_global_load_async_to_lds_b128(
        (AS1 v4i*)(v4i*)(uintptr_t)g, (AS3 v4i*)(v4i*)l, 0, 0);
#else
    *(v4f*)l = *(const v4f*)g;      // fallback: VGPR round-trip copy
#endif
}

// Wait until at most N async-LDS loads remain outstanding (in-order completion).
static __device__ __forceinline__ void wait_async_lds_all() {
#if HAS_ASYNC_LDS
#if __has_builtin(__builtin_amdgcn_s_wait_asynccnt)
    __builtin_amdgcn_s_wait_asynccnt(0);
#else
    asm volatile("s_wait_asynccnt 0x0" ::: "memory");
#endif
#endif
}
static __device__ __forceinline__ void wait_async_lds_prev() {
#if HAS_ASYNC_LDS
#if __has_builtin(__builtin_amdgcn_s_wait_asynccnt)
    __builtin_amdgcn_s_wait_asynccnt(LOADS_PER_THREAD);
#else
    asm volatile("s_wait_asynccnt 0x4" ::: "memory");   // == LOADS_PER_THREAD
#endif
#endif
}

// D = A(16x4 f32) * B(4x16 f32) + C(16x16 f32), full fp32 matrix pipe (CDNA5 WMMA)
static __device__ __forceinline__ v8f wmma_f32(v2f a, v2f b, v8f c) {
    return __builtin_amdgcn_wmma_f32_16x16x4_f32(
        /*neg_a=*/false, a, /*neg_b=*/false, b,
        /*c_mod=*/(short)0, c, /*reuse_a=*/false, /*reuse_b=*/false);
}

// ---------------------------------------------------------------------------
// Kernel 1: H[r, o] = sum_f X[r, f] * W[o, f]   (16384 x 256, K=256)
// Block: 256 thr = 8 waves; 16-row tile x 256 cols; wave = 16x32 via 2 WMMA accs.
// ---------------------------------------------------------------------------
__global__ __launch_bounds__(256) void k_gemm1(const float* __restrict__ X,
                                               const float* __restrict__ W,
                                               float* __restrict__ H) {
    const int wave = threadIdx.x >> 5;
    const int lane = threadIdx.x & 31;
    const int lrow = lane & 15;
    const int kb   = (lane >> 4) << 1;
    const int rowTile = blockIdx.x << 4;
    const int col0    = wave << 5;

    const float* xr = X + (size_t)(rowTile + lrow) * F_;
    const float* w0 = W + (size_t)(col0 + lrow) * F_;
    const float* w1 = W + (size_t)(col0 + 16 + lrow) * F_;

    v8f acc0 = {}; v8f acc1 = {};
#pragma unroll 4
    for (int f = 0; f < F_; f += 4) {
        v2f a  = *(const v2f*)(xr + f + kb);
        v2f b0 = *(const v2f*)(w0 + f + kb);
        v2f b1 = *(const v2f*)(w1 + f + kb);
        acc0 = wmma_f32(a, b0, acc0);
        acc1 = wmma_f32(a, b1, acc1);
    }
    const int rofs = (lane >> 4) << 3;
#pragma unroll
    for (int r = 0; r < 8; ++r) {
        const size_t row = (size_t)(rowTile + r + rofs);
        H[row * F_ + col0 + lrow]      = acc0[r];
        H[row * F_ + col0 + 16 + lrow] = acc1[r];
    }
}

// ---------------------------------------------------------------------------
// Kernel 2: s1[r] = H[r,:].a1 ; s2[r] = H[r,:].a2   (one wave per row)
// ---------------------------------------------------------------------------
__global__ __launch_bounds__(256) void k_scores(const float* __restrict__ H,
                                                const float* __restrict__ a1,
                                                const float* __restrict__ a2,
                                                float* __restrict__ s1,
                                                float* __restrict__ s2) {
    const int wave = threadIdx.x >> 5;
    const int lane = threadIdx.x & 31;
    const int row  = (blockIdx.x << 3) + wave;
    const float* hr = H + (size_t)row * F_;
    const int o = lane << 3;

    v4f h0 = *(const v4f*)(hr + o);
    v4f h1 = *(const v4f*)(hr + o + 4);
    v4f p0 = *(const v4f*)(a1 + o);
    v4f p1 = *(const v4f*)(a1 + o + 4);
    v4f q0 = *(const v4f*)(a2 + o);
    v4f q1 = *(const v4f*)(a2 + o + 4);

    float d1 = h0.x*p0.x + h0.y*p0.y + h0.z*p0.z + h0.w*p0.w
             + h1.x*p1.x + h1.y*p1.y + h1.z*p1.z + h1.w*p1.w;
    float d2 = h0.x*q0.x + h0.y*q0.y + h0.z*q0.z + h0.w*q0.w
             + h1.x*q1.x + h1.y*q1.y + h1.z*q1.z + h1.w*q1.w;

#pragma unroll
    for (int m = 16; m >= 1; m >>= 1) {
        d1 += __shfl_xor(d1, m, 32);
        d2 += __shfl_xor(d2, m, 32);
    }
    if (lane == 0) { s1[row] = d1; s2[row] = d2; }
}

// ---------------------------------------------------------------------------
// Kernel 3: M2[b] = max_m s2[b, m]
// ---------------------------------------------------------------------------
__global__ __launch_bounds__(1024) void k_maxs2(const float* __restrict__ s2,
                                                float* __restrict__ M2) {
    __shared__ float red[32];
    const int b   = blockIdx.x;
    const int tid = threadIdx.x;
    float v = -3.4e38f;
    for (int i = tid; i < N_; i += 1024) v = fmaxf(v, s2[b * N_ + i]);
#pragma unroll
    for (int m = 16; m >= 1; m >>= 1) v = fmaxf(v, __shfl_xor(v, m, 32));
    if ((tid & 31) == 0) red[tid >> 5] = v;
    __syncthreads();
    if (tid < 32) {
        v = red[tid];
#pragma unroll
        for (int m = 16; m >= 1; m >>= 1) v = fmaxf(v, __shfl_xor(v, m, 32));
        if (tid == 0) M2[b] = v;
    }
}

// ---------------------------------------------------------------------------
// Kernel 4: fused flash-style attention, double-buffered async-LDS pipeline.
// att[n,m] = LReLU(s1[n]+s2[m]); rowmax = LReLU(s1[n]+M2) (LReLU monotone).
// Chunk c+1 is async-copied into the alternate LDS buffer while chunk c feeds
// the fp32 WMMAs; s_wait_asynccnt <= LOADS_PER_THREAD releases only chunk c
// (async loads complete in order). out = ALPHA*H + (1-ALPHA)*(P@H)/Z.
// Grid (N/32, B); block 256 = 8 waves = rowSub{0,1} x colGroup{0..3} (16x64 each).
// ---------------------------------------------------------------------------
__global__ __launch_bounds__(256) void k_attn(const float* __restrict__ H,
                                              const float* __restrict__ s1,
                                              const float* __restrict__ s2,
                                              const float* __restrict__ M2,
                                              float* __restrict__ out) {
    __shared__ __align__(16) float ldsH[2][CHUNK_M * F_];   // 2 x 16 KB

    const int b    = blockIdx.y;
    const int tid  = threadIdx.x;
    const int wave = tid >> 5;
    const int lane = tid & 31;
    const int rowSub = wave & 1;
    const int colG   = wave >> 1;
    const int lrow = lane & 15;
    const int kb   = (lane >> 4) << 1;
    const int rbase = (blockIdx.x << 5) + (rowSub << 4);

    const float* Hb  = H  + (size_t)b * N_ * F_;
    const float* s1b = s1 + b * N_;
    const float* s2b = s2 + b * N_;

    const float s1v  = s1b[rbase + lrow];
    const float tm   = s1v + M2[b];
    const float rmax = (tm >= 0.f) ? tm : BETA * tm;

    v8f acc0 = {}, acc1 = {}, acc2 = {}, acc3 = {};
    float psum = 0.f;
    const int c0 = (colG << 6) + lrow;

    // ---- prologue: fill buffer 0 with chunk 0
#pragma unroll
    for (int t = 0; t < LOADS_PER_THREAD; ++t) {
        const int off = (t << 10) + (tid << 2);
        async_g2l_b128(Hb + off, &ldsH[0][off]);
    }

#pragma unroll 1
    for (int c = 0; c < NCHUNK; ++c) {
        const int m0  = c * CHUNK_M;
        const int cur = c & 1;

        if (c + 1 < NCHUNK) {
            // issue next chunk into the alternate buffer, then release current
            const float* gsrc = Hb + (size_t)(m0 + CHUNK_M) * F_;
#pragma unroll
            for (int t = 0; t < LOADS_PER_THREAD; ++t) {
                const int off = (t << 10) + (tid << 2);
                async_g2l_b128(gsrc + off, &ldsH[cur ^ 1][off]);
            }
            wait_async_lds_prev();     // chunk c landed; chunk c+1 still streaming
        } else {
            wait_async_lds_all();
        }
        __syncthreads();

        // ---- 4 K-steps of 4 from the current LDS buffer
        const float* lbuf = &ldsH[cur][0];
#pragma unroll
        for (int st = 0; st < CHUNK_M / 4; ++st) {
            const int mi = (st << 2) + kb;               // local m row: kb / kb+1
            v2f sv = *(const v2f*)(s2b + m0 + mi);
            float t0 = s1v + sv.x; t0 = (t0 >= 0.f) ? t0 : BETA * t0;
            float t1 = s1v + sv.y; t1 = (t1 >= 0.f) ? t1 : BETA * t1;
            const float e0 = __expf(t0 - rmax);
            const float e1 = __expf(t1 - rmax);
            psum += e0 + e1;
            v2f a; a.x = e0; a.y = e1;

            const float* lr = lbuf + mi * F_ + c0;
            v2f b0; b0.x = lr[0];   b0.y = lr[F_];
            v2f b1; b1.x = lr[16];  b1.y = lr[16 + F_];
            v2f b2; b2.x = lr[32];  b2.y = lr[32 + F_];
            v2f b3; b3.x = lr[48];  b3.y = lr[48 + F_];

            acc0 = wmma_f32(a, b0, acc0);
            acc1 = wmma_f32(a, b1, acc1);
            acc2 = wmma_f32(a, b2, acc2);
            acc3 = wmma_f32(a, b3, acc3);
        }
        __syncthreads();               // current buffer free for chunk c+2
    }

    psum += __shfl_xor(psum, 16, 32);  // lane L now holds Z for row (L&15)

    const int rofs = (lane >> 4) << 3;
#pragma unroll
    for (int r = 0; r < 8; ++r) {
        const int row     = rbase + r + rofs;
        const float z     = __shfl(psum, r + rofs, 32);
        const float scale = (1.0f - ALPHA) / z;          // Z >= 1 (max attained)
        const size_t hrow = (size_t)row * F_;
        const size_t obase = ((size_t)b * N_ + row) * F_;
        out[obase + c0]      = ALPHA * Hb[hrow + c0]      + scale * acc0[r];
        out[obase + c0 + 16] = ALPHA * Hb[hrow + c0 + 16] + scale * acc1[r];
        out[obase + c0 + 32] = ALPHA * Hb[hrow + c0 + 32] + scale * acc2[r];
        out[obase + c0 + 48] = ALPHA * Hb[hrow + c0 + 48] + scale * acc3[r];
    }
}

// ---------------------------------------------------------------------------
extern "C" void kernel_launch(void* const* d_in, const int* in_sizes, int n_in,
                              void* d_out, int out_size, void* d_ws, size_t ws_size,
                              hipStream_t stream) {
    const float* x  = (const float*)d_in[0];   // [4, 4096, 256]
    const float* W  = (const float*)d_in[1];   // [256, 256] (out, in)
    const float* a1 = (const float*)d_in[2];   // [256]
    const float* a2 = (const float*)d_in[3];   // [256]
    float* out = (float*)d_out;                // [4, 4096, 256]

    char* ws = (char*)d_ws;
    float* H  = (float*)ws;                                  // 16 MB
    float* s1 = (float*)(ws + (size_t)ROWS_TOTAL * F_ * 4);  // 64 KB
    float* s2 = s1 + ROWS_TOTAL;                             // 64 KB
    float* M2 = s2 + ROWS_TOTAL;                             // 16 B

    k_gemm1 <<<dim3(ROWS_TOTAL / 16), dim3(256), 0, stream>>>(x, W, H);
    k_scores<<<dim3(ROWS_TOTAL / 8),  dim3(256), 0, stream>>>(H, a1, a2, s1, s2);
    k_maxs2 <<<dim3(B_),              dim3(1024), 0, stream>>>(s2, M2);
    k_attn  <<<dim3(N_ / 32, B_),     dim3(256), 0, stream>>>(H, s1, s2, M2, out);
}